// myAttention_15083925143810
// MI455X (gfx1250) — compile-verified
//
#include <hip/hip_runtime.h>

typedef float v2f __attribute__((ext_vector_type(2)));
typedef float v4f __attribute__((ext_vector_type(4)));
typedef float v8f __attribute__((ext_vector_type(8)));

#define WAVES_PER_BLOCK 8
#define THREADS (WAVES_PER_BLOCK * 32)
#define GPW 8   // batch-groups (of 4 batches) per wave

__global__ __launch_bounds__(THREADS)
void tokatt_wmma_kernel(const float* __restrict__ in0,
                        const float* __restrict__ in1,
                        const float* __restrict__ in2,
                        const float* __restrict__ in3,
                        float* __restrict__ out,   // [B,16]
                        float* __restrict__ att,   // [B,4,4]
                        long Btot)
{
    // per-wave att staging tile: [wave][batch-sub][i][j]
    __shared__ float latt[WAVES_PER_BLOCK][4][4][4];

    const int tid  = (int)threadIdx.x;
    const int lane = tid & 31;
    // wave index is uniform across the wave: pin it to an SGPR so the trip
    // count below is provably scalar -> uniform s_cmp/s_cbranch loop, no
    // per-iteration exec-mask bookkeeping.
    const int wave = __builtin_amdgcn_readfirstlane(tid >> 5);
    const int half = lane >> 4;   // K-pair half (also D-matrix row-half)
    const int l16  = lane & 15;
    const int bs   = l16 >> 2;    // batch-sub 0..3 within the 4-batch tile
    const int tok  = lane & 3;    // token slot (selects input array); also j on useful lanes

    // lane's input array: token slot == which mlp input
    const float* src = (tok == 0) ? in0 : (tok == 1) ? in1 : (tok == 2) ? in2 : in3;

    // lanes whose D-registers hold a diagonal (same-batch) Gram column
    const bool useful = (half == (bs >> 1));

    // wave-uniform scalar trip count
    const long waveGlobal  = (long)blockIdx.x * WAVES_PER_BLOCK + wave;
    const long totalGroups = (Btot + 3) >> 2;
    const long grp0        = waveGlobal * GPW;
    long nIt = totalGroups - grp0;
    nIt = (nIt < 0) ? 0 : (nIt > GPW ? GPW : nIt);

    // per-lane streaming pointers; each iteration advances one group = 4 batches = 64 floats
    const long bBase = grp0 * 4 + bs;
    const float* pin  = src + bBase * 16 + 2 * half;
    float*       patt = att + bBase * 16 + tok * 4;
    float*       pout = out + bBase * 16 + 4 * tok + 2 * half;

    for (long g = 0; g < nIt; ++g) {
        // lane covers d = {4*dc + 2*half, +1} for dc=0..3  (8 floats, b64 NT loads:
        // zero-reuse stream, don't churn L2)
        const v2f x0 = __builtin_nontemporal_load((const v2f*)(pin + 0));
        const v2f x1 = __builtin_nontemporal_load((const v2f*)(pin + 4));
        const v2f x2 = __builtin_nontemporal_load((const v2f*)(pin + 8));
        const v2f x3 = __builtin_nontemporal_load((const v2f*)(pin + 12));

        // D = A * A^T accumulated over 4 K-chunks: diagonal 4x4 blocks = per-batch Grams.
        // f32 A(16x4) and B(4x16) lane layouts mirror, so the same regs feed both operands.
        v8f acc = {0.f, 0.f, 0.f, 0.f, 0.f, 0.f, 0.f, 0.f};
        acc = __builtin_amdgcn_wmma_f32_16x16x4_f32(false, x0, false, x0, (short)0, acc, false, false);
        acc = __builtin_amdgcn_wmma_f32_16x16x4_f32(false, x1, false, x1, (short)0, acc, false, false);
        acc = __builtin_amdgcn_wmma_f32_16x16x4_f32(false, x2, false, x2, (short)0, acc, false, false);
        acc = __builtin_amdgcn_wmma_f32_16x16x4_f32(false, x3, false, x3, (short)0, acc, false, false);

        // Useful lane holds Gram column j=tok of batch b: rows in acc[r0..r0+3], r0 = 4*(bs&1).
        // softmax over axis=1 (rows i) is within-lane.
        if (useful) {
            const bool hi = (bs & 1);
            const float c0 = hi ? acc[4] : acc[0];
            const float c1 = hi ? acc[5] : acc[1];
            const float c2 = hi ? acc[6] : acc[2];
            const float c3 = hi ? acc[7] : acc[3];
            const float m  = fmaxf(fmaxf(c0, c1), fmaxf(c2, c3));
            const float e0 = __expf(c0 - m);
            const float e1 = __expf(c1 - m);
            const float e2 = __expf(c2 - m);
            const float e3 = __expf(c3 - m);
            // softmax normalizer: single v_rcp_f32 instead of IEEE divide chain
            const float inv = __builtin_amdgcn_rcpf(e0 + e1 + e2 + e3);
            latt[wave][bs][0][tok] = e0 * inv;
            latt[wave][bs][1][tok] = e1 * inv;
            latt[wave][bs][2][tok] = e2 * inv;
            latt[wave][bs][3][tok] = e3 * inv;
        }
        // same-wave producer/consumer: DS ops are in-order per wave; just drain DScnt
        asm volatile("s_wait_dscnt 0x0" ::: "memory");

        // row i=tok of this batch's att tile (contiguous 16B in LDS)
        const v4f row = *(const v4f*)&latt[wave][bs][tok][0];

        // att out: useful lanes store one row each -> 4 x b128 = full 64B tile per batch
        if (useful) {
            __builtin_nontemporal_store(row, (v4f*)patt);
        }

        // w[i] = row-sum of att (columns are softmax-normalized, rows are not)
        const float wt = row.x + row.y + row.z + row.w;

        // partial output: x[d,tok] * w[tok] for this lane's 8 d-values
        v2f s0 = { x0.x * wt, x0.y * wt };
        v2f s1 = { x1.x * wt, x1.y * wt };
        v2f s2 = { x2.x * wt, x2.y * wt };
        v2f s3 = { x3.x * wt, x3.y * wt };

        // stage 1: fold lanes {L, L^2} for all four d-chunks
        s0.x += __shfl_xor(s0.x, 2); s0.y += __shfl_xor(s0.y, 2);
        s1.x += __shfl_xor(s1.x, 2); s1.y += __shfl_xor(s1.y, 2);
        s2.x += __shfl_xor(s2.x, 2); s2.y += __shfl_xor(s2.y, 2);
        s3.x += __shfl_xor(s3.x, 2); s3.y += __shfl_xor(s3.y, 2);

        // stage 2: keep my chunk s[tok]; hand partner its chunk s[tok^1]; one xor-1 exchange
        const bool b0sel = (tok & 1);
        const bool b1sel = (tok & 2);
        const v2f a   = b0sel ? s1 : s0;   // s[b0]
        const v2f bq  = b0sel ? s0 : s1;   // s[b0^1]
        const v2f c   = b0sel ? s3 : s2;   // s[2+b0]
        const v2f dq  = b0sel ? s2 : s3;   // s[2+(b0^1)]
        v2f keep = b1sel ? c  : a;         // s[tok]
        v2f send = b1sel ? dq : bq;        // s[tok^1]
        keep.x += __shfl_xor(send.x, 1);
        keep.y += __shfl_xor(send.y, 1);

        // lane writes the d-chunk dc == tok: out[b, 4*tok + 2*half .. +1]  (b64 store)
        __builtin_nontemporal_store(keep, (v2f*)pout);

        pin  += 64;
        patt += 64;
        pout += 64;
    }
}

extern "C" void kernel_launch(void* const* d_in, const int* in_sizes, int n_in,
                              void* d_out, int out_size, void* d_ws, size_t ws_size,
                              hipStream_t stream) {
    const float* in0 = (const float*)d_in[0];
    const float* in1 = (const float*)d_in[1];
    const float* in2 = (const float*)d_in[2];
    const float* in3 = (const float*)d_in[3];
    const long Btot = (long)in_sizes[0] / 16;           // [B, 16] inputs
    float* out = (float*)d_out;                          // first output: [B,16]
    float* att = out + Btot * 16;                        // second output: [B,4,4]

    const long groups = (Btot + 3) / 4;
    const long waves  = (groups + GPW - 1) / GPW;
    const long blocks = (waves + WAVES_PER_BLOCK - 1) / WAVES_PER_BLOCK;

    tokatt_wmma_kernel<<<(unsigned)blocks, THREADS, 0, stream>>>(
        in0, in1, in2, in3, out, att, Btot);
}